// LSTMDecoderAttention_3504693314118
// MI455X (gfx1250) — compile-verified
//
#include <hip/hip_runtime.h>
#include <hip/hip_bf16.h>
#include <math.h>

#define B_SZ  256
#define TC_SZ 256
#define D_SZ  2048

typedef __attribute__((ext_vector_type(2))) float v2f;
typedef __attribute__((ext_vector_type(4))) float v4f;
typedef __attribute__((ext_vector_type(8))) float v8f;

// ---------------------------------------------------------------------------
// Kernel 1: hw[b,n] = sum_k h_dec[b,k] * W[k,n] via V_WMMA_F32_16X16X4_F32.
// One wave per 16x16 output tile; 4 waves per block cover 16(M) x 64(N).
// A 16x4 layout: lanes 0-15 -> M=lane, VGPR0/1 = K0/K1; lanes 16-31 -> K2/K3.
// B 4x16 layout: N = lane&15, lane-half selects K pair, VGPR0/1 = k, k+1.
// ---------------------------------------------------------------------------
__global__ __launch_bounds__(128) void gemm_hw_wmma(const float* __restrict__ h_dec,
                                                    const float* __restrict__ W,
                                                    float* __restrict__ hw) {
    const int lane = threadIdx.x & 31;
    const int wave = threadIdx.x >> 5;
    const int bm   = blockIdx.y;                  // 0..B/16-1
    const int n0   = blockIdx.x * 64 + wave * 16; // N tile base
    const int lh   = lane & 15;                   // M (for A) / N (for B,C)
    const int ls   = lane >> 4;                   // half-wave selector

    const float* Ap = h_dec + (size_t)(bm * 16 + lh) * D_SZ + 2 * ls;
    const float* Bp = W + (size_t)(2 * ls) * D_SZ + n0 + lh;

    v8f c = {0.f, 0.f, 0.f, 0.f, 0.f, 0.f, 0.f, 0.f};
#pragma unroll 4
    for (int k = 0; k < D_SZ; k += 4) {
        v2f a = *(const v2f*)Ap;   // {A[m,k+2ls], A[m,k+2ls+1]} : 8B aligned
        v2f b;
        b.x = Bp[0];               // W[k+2ls  , n0+lh]
        b.y = Bp[D_SZ];            // W[k+2ls+1, n0+lh]
        c = __builtin_amdgcn_wmma_f32_16x16x4_f32(
                /*neg_a=*/false, a, /*neg_b=*/false, b,
                /*c_mod=*/(short)0, c, /*reuse_a=*/false, /*reuse_b=*/false);
        Ap += 4;
        Bp += (size_t)4 * D_SZ;
    }

    // C/D layout: VGPR r -> M = r + 8*ls, N = n0 + lh
    const int nCol = n0 + lh;
#pragma unroll
    for (int r = 0; r < 8; ++r) {
        const int m = r + 8 * ls;
        hw[(size_t)(bm * 16 + m) * D_SZ + nCol] = c[r];
    }
}

// ---------------------------------------------------------------------------
// Kernel 2 (fused): stream prev_h_dec ONCE -> (a) copy row into new_prev_h_dec,
// (b) dot with hw[b,:] for the raw attention score.  One wave (32 lanes) per
// (b,t) row: each lane handles 16 float4 chunks (64 floats) of the 2048-row.
// ---------------------------------------------------------------------------
__global__ __launch_bounds__(256) void scores_and_copy(const float* __restrict__ hw,
                                                       const float* __restrict__ prev,
                                                       const unsigned char* __restrict__ mask,
                                                       float* __restrict__ newprev,
                                                       float* __restrict__ e_ws) {
    const int lane = threadIdx.x & 31;
    const int wave = threadIdx.x >> 5;                 // 0..7
    const int b    = blockIdx.x >> 5;                  // 32 blocks per batch row
    const int t    = (blockIdx.x & 31) * 8 + wave;     // 0..TC-1

    const float* src  = prev    + ((size_t)b * TC_SZ + t) * D_SZ;
    float*       dst  = newprev + ((size_t)b * (TC_SZ + 1) + t) * D_SZ;
    const float* hrow = hw      + (size_t)b * D_SZ;    // 8 KB, reused by 256 t's -> hot in cache

    float acc = 0.f;
#pragma unroll
    for (int i = 0; i < 16; ++i) {
        const int d = i * 128 + lane * 4;
        v4f p = *(const v4f*)(src + d);
        v4f h = *(const v4f*)(hrow + d);
        *(v4f*)(dst + d) = p;                          // fused concat copy
        acc += p.x * h.x + p.y * h.y + p.z * h.z + p.w * h.w;
    }
    // wave32 butterfly reduction
#pragma unroll
    for (int off = 16; off >= 1; off >>= 1)
        acc += __shfl_xor(acc, off, 32);

    if (lane == 0) {
        const bool m = mask[(size_t)b * TC_SZ + t] != 0;
        e_ws[(size_t)b * TC_SZ + t] = m ? acc : -INFINITY;
    }
}

// ---------------------------------------------------------------------------
// Kernel 3: append h_dec as row TC of new_prev_h_dec (float4 copy, 2 MB).
// ---------------------------------------------------------------------------
__global__ __launch_bounds__(256) void append_hdec(const float* __restrict__ h_dec,
                                                   float* __restrict__ newprev) {
    const size_t gid = (size_t)blockIdx.x * blockDim.x + threadIdx.x;  // float4 idx
    const int d4 = D_SZ / 4;
    const int bb = (int)(gid / d4);
    const int r  = (int)(gid % d4);
    const v4f* src = (const v4f*)h_dec;
    v4f*       dst = (v4f*)newprev;
    dst[(size_t)bb * ((TC_SZ + 1) * d4) + (size_t)TC_SZ * d4 + r] = src[gid];
}

// ---------------------------------------------------------------------------
// Kernel 4: max-stabilized softmax over TC=256 per batch row (in-place in ws).
// ---------------------------------------------------------------------------
__global__ __launch_bounds__(256) void softmax_rows(float* __restrict__ e_ws) {
    __shared__ float red[256];
    const int b = blockIdx.x;
    const int t = threadIdx.x;
    const float v = e_ws[(size_t)b * TC_SZ + t];

    red[t] = v;
    __syncthreads();
#pragma unroll
    for (int s = 128; s > 0; s >>= 1) {
        if (t < s) red[t] = fmaxf(red[t], red[t + s]);
        __syncthreads();
    }
    const float mx = red[0];
    __syncthreads();

    const float ex = __expf(v - mx);   // exp(-inf - mx) -> 0 for masked slots
    red[t] = ex;
    __syncthreads();
#pragma unroll
    for (int s = 128; s > 0; s >>= 1) {
        if (t < s) red[t] += red[t + s];
        __syncthreads();
    }
    e_ws[(size_t)b * TC_SZ + t] = ex / red[0];
}

// ---------------------------------------------------------------------------
// Kernel 5: context[b,d] = sum_t w[b,t] * prev[b,t,d].  Threads span d
// (coalesced per t); w[b,t] is wave-uniform -> scalar loads.
// ---------------------------------------------------------------------------
__global__ __launch_bounds__(256) void context_kernel(const float* __restrict__ w_ws,
                                                      const float* __restrict__ prev,
                                                      float* __restrict__ ctx) {
    const int b = blockIdx.y;
    const int d = blockIdx.x * 256 + threadIdx.x;
    const float* p = prev + (size_t)b * TC_SZ * D_SZ + d;
    const float* w = w_ws + (size_t)b * TC_SZ;

    float acc = 0.f;
#pragma unroll 8
    for (int t = 0; t < TC_SZ; ++t)
        acc = fmaf(w[t], p[(size_t)t * D_SZ], acc);

    ctx[(size_t)b * D_SZ + d] = acc;
}

// ---------------------------------------------------------------------------
extern "C" void kernel_launch(void* const* d_in, const int* in_sizes, int n_in,
                              void* d_out, int out_size, void* d_ws, size_t ws_size,
                              hipStream_t stream) {
    const float*         h_dec = (const float*)d_in[0];
    const float*         prev  = (const float*)d_in[1];
    const unsigned char* mask  = (const unsigned char*)d_in[2];
    const float*         W     = (const float*)d_in[3];

    // Outputs, flat in return order: context [B,D] then new_prev [B,TC+1,D]
    float* ctx     = (float*)d_out;
    float* newprev = (float*)d_out + (size_t)B_SZ * D_SZ;

    // Workspace: hw [B,D] then e/weights [B,TC]  (~2.3 MB total)
    float* hw   = (float*)d_ws;
    float* e_ws = hw + (size_t)B_SZ * D_SZ;

    gemm_hw_wmma   <<<dim3(D_SZ / 64, B_SZ / 16), 128, 0, stream>>>(h_dec, W, hw);
    scores_and_copy<<<B_SZ * (TC_SZ / 8),          256, 0, stream>>>(hw, prev, mask, newprev, e_ws);
    append_hdec    <<<(B_SZ * D_SZ / 4) / 256,     256, 0, stream>>>(h_dec, newprev);
    softmax_rows   <<<B_SZ,                        256, 0, stream>>>(e_ws);
    context_kernel <<<dim3(D_SZ / 256, B_SZ),      256, 0, stream>>>(e_ws, prev, ctx);
}